// LengthRegulator_70394513981840
// MI455X (gfx1250) — compile-verified
//
#include <hip/hip_runtime.h>
#include <hip/hip_bf16.h>

#define B_    16
#define T_IN  512
#define D_    256
#define TOUT  4096
#define KW    3
#define KTOT  (D_ * KW)          // 768 im2col K

typedef __attribute__((ext_vector_type(2))) float v2f;
typedef __attribute__((ext_vector_type(8))) float v8f;

// ---------------------------------------------------------------------------
// Kernel 0: repack conv weights W[co][ci][k] into WMMA-native B layout:
//   Wp[ ((K/4)*256 + co)*4 + (K&3) ] = W[co][ci][k],  K = k*256 + ci
// so a wave's 16x4 B fragment is one contiguous 256B region (coalesced b64).
// ---------------------------------------------------------------------------
__global__ __launch_bounds__(256) void pack_weights(const float* __restrict__ W,
                                                    float* __restrict__ Wp) {
    const int idx = blockIdx.x * 256 + threadIdx.x;     // 0 .. 196607
    const int kv     = idx & 3;
    const int co     = (idx >> 2) & 255;
    const int kchunk = idx >> 10;
    const int K    = kchunk * 4 + kv;
    const int slab = K >> 8;           // kernel tap 0..2
    const int ci   = K & 255;
    Wp[idx] = W[co * KTOT + ci * KW + slab];
}

// ---------------------------------------------------------------------------
// Kernel 1: per-batch inclusive scan of target -> ends  (B blocks x 512 thr)
// ---------------------------------------------------------------------------
__global__ __launch_bounds__(512) void scan_kernel(const int* __restrict__ target,
                                                   int* __restrict__ ends) {
    __shared__ int buf[T_IN];
    const int b = blockIdx.x;
    const int tid = threadIdx.x;
    buf[tid] = target[b * T_IN + tid];
    __syncthreads();
    for (int off = 1; off < T_IN; off <<= 1) {
        int v = (tid >= off) ? buf[tid - off] : 0;
        __syncthreads();
        buf[tid] += v;
        __syncthreads();
    }
    ends[b * T_IN + tid] = buf[tid];
}

// ---------------------------------------------------------------------------
// Kernel 2: length-regulator expansion. out[b,t,:] = x[b, upper_bound(ends,t), :]
// or zeros past the total. Grid (TOUT/8, B), 256 threads: one wave per out row,
// each lane stores 32 bytes (2 x float4) -> fully coalesced 1KB row stores.
// ---------------------------------------------------------------------------
__global__ __launch_bounds__(256) void expand_kernel(const float* __restrict__ x,
                                                     const int* __restrict__ ends,
                                                     float* __restrict__ out) {
    __shared__ int e[T_IN];
    const int b = blockIdx.y;
    for (int i = threadIdx.x; i < T_IN; i += 256) e[i] = ends[b * T_IN + i];
    __syncthreads();
    const int total = e[T_IN - 1];
    const int lane = threadIdx.x & 31;
    const int t = blockIdx.x * 8 + (threadIdx.x >> 5);

    float4 v0 = make_float4(0.f, 0.f, 0.f, 0.f);
    float4 v1 = v0;
    if (t < total) {
        // first i with ends[i] > t  (empty intervals are skipped naturally)
        int lo = 0, hi = T_IN;
        while (lo < hi) {
            int mid = (lo + hi) >> 1;
            if (e[mid] > t) hi = mid; else lo = mid + 1;
        }
        const float4* src = (const float4*)(x + ((size_t)b * T_IN + lo) * D_);
        v0 = src[lane * 2 + 0];
        v1 = src[lane * 2 + 1];
    }
    float4* dst = (float4*)(out + ((size_t)b * TOUT + t) * D_);
    dst[lane * 2 + 0] = v0;
    dst[lane * 2 + 1] = v1;
}

// ---------------------------------------------------------------------------
// Kernel 3: conv1d(K=3, SAME, Cin=256, Cout=256) as im2col GEMM on
// V_WMMA_F32_16X16X4_F32, fused bias + LayerNorm + ReLU.
// One block per 16-row M tile (M = B*T = 8192 -> 512 blocks), 256 thr = 8 waves.
// Wave w owns output columns [w*32, w*32+32) via two 16x16 accumulators.
// B operand comes from the packed layout (coalesced v2f/b64 loads).
// ---------------------------------------------------------------------------
__global__ __launch_bounds__(256) void conv_ln_relu_wmma(const float* __restrict__ X,
                                                         const float* __restrict__ Wp,
                                                         const float* __restrict__ bias,
                                                         const float* __restrict__ g,
                                                         const float* __restrict__ be,
                                                         float* __restrict__ out) {
    __shared__ float xwin[18 * 256];   // 16 rows + 1-row halo each side
    __shared__ float ctile[16 * 256];  // fp32 result tile before LN
    __shared__ float redS[256];
    __shared__ float redQ[256];
    __shared__ float rowMean[16];
    __shared__ float rowInv[16];

    const int m0 = blockIdx.x * 16;     // global first row of tile
    const int b  = m0 / T_IN;
    const int t0 = m0 % T_IN;           // tiles never straddle batches (512 % 16 == 0)
    const int tid = threadIdx.x;

    // Stage halo window, zero-padded at sequence (= batch) boundaries.
    for (int idx = tid; idx < 18 * 256; idx += 256) {
        const int w = idx >> 8;         // 0..17  -> t = t0 - 1 + w
        const int c = idx & 255;
        const int t = t0 - 1 + w;
        float v = 0.f;
        if (t >= 0 && t < T_IN) v = X[((size_t)b * T_IN + t) * 256 + c];
        xwin[idx] = v;
    }
    __syncthreads();

    const int lane = tid & 31;
    const int wv   = tid >> 5;          // wave 0..7
    const int mA   = lane & 15;         // A-matrix row (M) held by this lane
    const int kb   = (lane >> 4) << 1;  // K sub-offset: lanes 16..31 hold K+2,K+3
    const int n0   = wv * 32;           // wave's N base
    const int nl   = lane & 15;         // B/C column within 16-wide tile

    v8f c0 = {};
    v8f c1 = {};

    // K = 3 slabs (kernel taps) x 256 channels; 4 K-values per WMMA issue.
    for (int k = 0; k < KW; ++k) {
        const float* xr  = &xwin[(mA + k) * 256 + kb];   // A[m][k*256 + kc + kb]
        // packed-B base for this slab: kchunk = k*64 + kc/4
        const float* wp0 = Wp + ((size_t)(k * 64) * 256 + n0 + nl) * 4 + kb;
        const float* wp1 = wp0 + 64;                      // +16 columns
#pragma unroll 4
        for (int kc = 0; kc < 256; kc += 4) {
            v2f a;
            a.x = xr[kc];
            a.y = xr[kc + 1];
            const v2f b0 = *(const v2f*)(wp0 + (kc << 8));  // (kc/4)*1024 floats
            const v2f b1 = *(const v2f*)(wp1 + (kc << 8));
            c0 = __builtin_amdgcn_wmma_f32_16x16x4_f32(false, a, false, b0,
                                                       (short)0, c0, false, false);
            c1 = __builtin_amdgcn_wmma_f32_16x16x4_f32(false, a, false, b1,
                                                       (short)0, c1, false, false);
        }
    }

    // C/D layout: VGPR r -> row r (lanes 0-15) / row r+8 (lanes 16-31), col = lane&15.
    const int chi = lane >> 4;
#pragma unroll
    for (int r = 0; r < 8; ++r) {
        const int row = r + chi * 8;
        ctile[row * 256 + n0 + nl]      = c0[r] + bias[n0 + nl];
        ctile[row * 256 + n0 + 16 + nl] = c1[r] + bias[n0 + 16 + nl];
    }
    __syncthreads();

    // LayerNorm over 256 channels per row (biased var, eps=1e-5) + ReLU.
    const int row = tid >> 4;   // 16 rows x 16 threads each
    const int sub = tid & 15;
    float s = 0.f, q = 0.f;
    for (int c = sub; c < 256; c += 16) {
        const float v = ctile[row * 256 + c];
        s += v;
        q += v * v;
    }
    redS[tid] = s;
    redQ[tid] = q;
    __syncthreads();
    if (sub == 0) {
        float S = 0.f, Q = 0.f;
        for (int j = 0; j < 16; ++j) {
            S += redS[row * 16 + j];
            Q += redQ[row * 16 + j];
        }
        const float mean = S * (1.f / 256.f);
        const float var  = Q * (1.f / 256.f) - mean * mean;
        rowMean[row] = mean;
        rowInv[row]  = rsqrtf(var + 1e-5f);
    }
    __syncthreads();
    const float mean = rowMean[row];
    const float inv  = rowInv[row];
    for (int c = sub; c < 256; c += 16) {
        const float v = (ctile[row * 256 + c] - mean) * inv * g[c] + be[c];
        out[((size_t)m0 + row) * 256 + c] = fmaxf(v, 0.f);
    }
}

// ---------------------------------------------------------------------------
// Kernel 4: final linear 256 -> 1 + ReLU. One wave per row.
// ---------------------------------------------------------------------------
__global__ __launch_bounds__(256) void linear_relu(const float* __restrict__ h,
                                                   const float* __restrict__ wl,
                                                   const float* __restrict__ bl,
                                                   float* __restrict__ dur) {
    const int row  = blockIdx.x * 8 + (threadIdx.x >> 5);
    const int lane = threadIdx.x & 31;
    const float* hr = h + (size_t)row * 256;
    float s = 0.f;
    for (int c = lane; c < 256; c += 32) s += hr[c] * wl[c];
#pragma unroll
    for (int off = 16; off; off >>= 1) s += __shfl_xor(s, off, 32);
    if (lane == 0) dur[row] = fmaxf(s + bl[0], 0.f);
}

// ---------------------------------------------------------------------------
extern "C" void kernel_launch(void* const* d_in, const int* in_sizes, int n_in,
                              void* d_out, int out_size, void* d_ws, size_t ws_size,
                              hipStream_t stream) {
    const float* x      = (const float*)d_in[0];
    const int*   target = (const int*)d_in[1];
    // d_in[2] = mel_max_length scalar (fixed 4096, matches TOUT)
    const float* w1  = (const float*)d_in[3];
    const float* b1  = (const float*)d_in[4];
    const float* g1  = (const float*)d_in[5];
    const float* be1 = (const float*)d_in[6];
    const float* w2  = (const float*)d_in[7];
    const float* b2  = (const float*)d_in[8];
    const float* g2  = (const float*)d_in[9];
    const float* be2 = (const float*)d_in[10];
    const float* wl  = (const float*)d_in[11];
    const float* bl  = (const float*)d_in[12];

    float* out = (float*)d_out;                      // [B, TOUT, D]
    float* dur = out + (size_t)B_ * TOUT * D_;       // [B, T_IN]

    char*  ws   = (char*)d_ws;
    int*   ends = (int*)ws;                          // B*T_IN ints (32 KB)
    float* wp1  = (float*)(ws + 32 * 1024);          // 768 KB packed conv1 weights
    float* wp2  = wp1 + (size_t)KTOT * D_;           // 768 KB packed conv2 weights
    float* h1   = wp2 + (size_t)KTOT * D_;           // [B*T_IN, 256]  8 MB
    float* h2   = h1 + (size_t)B_ * T_IN * D_;       // [B*T_IN, 256]  8 MB

    const int wElems = KTOT * D_;                    // 196608
    pack_weights<<<wElems / 256, 256, 0, stream>>>(w1, wp1);
    pack_weights<<<wElems / 256, 256, 0, stream>>>(w2, wp2);

    scan_kernel<<<B_, 512, 0, stream>>>(target, ends);
    expand_kernel<<<dim3(TOUT / 8, B_), 256, 0, stream>>>(x, ends, out);

    const int mtiles = (B_ * T_IN) / 16;             // 512
    conv_ln_relu_wmma<<<mtiles, 256, 0, stream>>>(x,  wp1, b1, g1, be1, h1);
    conv_ln_relu_wmma<<<mtiles, 256, 0, stream>>>(h1, wp2, b2, g2, be2, h2);

    linear_relu<<<(B_ * T_IN) / 8, 256, 0, stream>>>(h2, wl, bl, dur);
}